// FacenetLoss_12395275616920
// MI455X (gfx1250) — compile-verified
//
#include <hip/hip_runtime.h>
#include <hip/hip_bf16.h>

typedef __attribute__((ext_vector_type(2))) float v2f;
typedef __attribute__((ext_vector_type(8))) float v8f;

static constexpr int   kN      = 384;
static constexpr int   kD      = 128;
static constexpr int   kNT     = kN / 16;   // 24 tile blocks per dim
static constexpr float kMargin = 0.2f;
static constexpr float kRho    = 10.0f;

// ---------------------------------------------------------------------------
// sq[i] = ||E[i]||^2
// ---------------------------------------------------------------------------
__global__ void sq_kernel(const float* __restrict__ E, float* __restrict__ sq) {
    int i = blockIdx.x * blockDim.x + threadIdx.x;
    if (i < kN) {
        float s = 0.0f;
#pragma unroll 8
        for (int d = 0; d < kD; ++d) {
            float v = E[i * kD + d];
            s = fmaf(v, v, s);
        }
        sq[i] = s;
    }
}

// ---------------------------------------------------------------------------
// One wave per 16x16 tile of G = E*E^T via chained V_WMMA_F32_16X16X4_F32,
// fused with conversion to clamped squared distances.
// 576 tiles -> 72 blocks x 8 waves (wave32).
// ---------------------------------------------------------------------------
__global__ void __launch_bounds__(256)
gram_dist_kernel(const float* __restrict__ E,
                 const float* __restrict__ sq,
                 float* __restrict__ dist) {
    const int wave = threadIdx.x >> 5;
    const int lane = threadIdx.x & 31;
    const int tile = blockIdx.x * 8 + wave;    // 0..575, exact cover
    const int ib   = tile / kNT;
    const int jb   = tile % kNT;
    const int m    = lane & 15;
    const int half = lane >> 4;

    // A tile 16x4 f32 layout: lane(m,half) holds A[m][2*half], A[m][2*half+1]
    // B tile 4x16 f32 (B = E_jblock^T): same per-lane float2 pattern from E.
    const float* rowA = E + (ib * 16 + m) * kD + 2 * half;
    const float* rowB = E + (jb * 16 + m) * kD + 2 * half;

    v8f c = {};
#pragma unroll
    for (int kk = 0; kk < 32; ++kk) {          // K = 128 = 32 * 4
        v2f a = *(const v2f*)(rowA + kk * 4);
        v2f b = *(const v2f*)(rowB + kk * 4);
        c = __builtin_amdgcn_wmma_f32_16x16x4_f32(
                /*neg_a=*/false, a, /*neg_b=*/false, b,
                /*c_mod=*/(short)0, c, /*reuse_a=*/false, /*reuse_b=*/false);
    }

    // C/D layout: VGPR g -> row (g + 8*half), col (lane&15)
    const int   n   = lane & 15;
    const float sqj = sq[jb * 16 + n];
#pragma unroll
    for (int g = 0; g < 8; ++g) {
        const int   gi = ib * 16 + g + 8 * half;
        const int   gj = jb * 16 + n;
        const float dv = sq[gi] + sqj - 2.0f * c[g];
        dist[gi * kN + gj] = fmaxf(dv, 0.0f);
    }
}

// ---------------------------------------------------------------------------
// Triplet scan: one block (4 wave32) per anchor i. Row of distances and all
// classes staged in LDS. Each wave owns a subset of positives j; lanes stride
// over k with butterfly max/min reductions.
// ---------------------------------------------------------------------------
__global__ void __launch_bounds__(128)
loss_kernel(const float* __restrict__ dist,
            const int*   __restrict__ classes,
            float*       __restrict__ acc) {
    __shared__ float sdist[kN];
    __shared__ int   scls[kN];

    const int i   = blockIdx.x;
    const int tid = threadIdx.x;

    for (int t = tid; t < kN; t += 128) {
        sdist[t] = dist[i * kN + t];
        scls[t]  = classes[t];
    }
    __syncthreads();

    const int ci   = scls[i];
    const int wave = tid >> 5;
    const int lane = tid & 31;

    float numLocal = 0.0f;
    float denLocal = 0.0f;

    for (int j = wave; j < kN; j += 4) {       // uniform per-wave branch
        if (scls[j] != ci || j == i) continue; // j must be a positive of i

        const float dij = sdist[j];
        float semi    = 0.0f;                  // max of semihard losses
        float shifted = 3.0e38f;               // min of (L - RHO*hard)

        for (int k = lane; k < kN; k += 32) {
            const float L      = fmaxf(dij - sdist[k] + kMargin, 0.0f);
            const bool  validk = (scls[k] != ci);
            const float s      = (validk && L > 0.0f && L <= kMargin) ? L : 0.0f;
            semi = fmaxf(semi, s);
            const float sh = L - ((validk && L > kMargin) ? kRho : 0.0f);
            shifted = fminf(shifted, sh);
        }
#pragma unroll
        for (int off = 16; off >= 1; off >>= 1) {
            semi    = fmaxf(semi,    __shfl_xor(semi,    off, 32));
            shifted = fminf(shifted, __shfl_xor(shifted, off, 32));
        }

        float pp = semi;
        if (pp == 0.0f && shifted < 0.0f) pp = shifted + kRho;
        numLocal += pp;
        denLocal += 1.0f;
    }

    if (lane == 0) {
        atomicAdd(&acc[0], numLocal);
        atomicAdd(&acc[1], denLocal);
    }
}

__global__ void init_kernel(float* __restrict__ acc) {
    acc[threadIdx.x] = 0.0f;
}

__global__ void finalize_kernel(const float* __restrict__ acc,
                                float* __restrict__ out) {
    const float num = acc[0];
    const float den = acc[1];
    out[0] = (den > 0.0f) ? (num / fmaxf(den, 1.0f)) : 0.0f;
}

// ---------------------------------------------------------------------------
extern "C" void kernel_launch(void* const* d_in, const int* in_sizes, int n_in,
                              void* d_out, int out_size, void* d_ws, size_t ws_size,
                              hipStream_t stream) {
    const int*   classes = (const int*)d_in[0];   // int32 [384]
    const float* E       = (const float*)d_in[1]; // f32   [384,128]
    float*       out     = (float*)d_out;

    float* dist = (float*)d_ws;       // 384*384 floats
    float* sq   = dist + kN * kN;     // 384 floats
    float* acc  = sq + kN;            // acc[0]=num, acc[1]=den

    init_kernel<<<1, 2, 0, stream>>>(acc);
    sq_kernel<<<(kN + 127) / 128, 128, 0, stream>>>(E, sq);
    gram_dist_kernel<<<(kNT * kNT) / 8, 256, 0, stream>>>(E, sq, dist);
    loss_kernel<<<kN, 128, 0, stream>>>(dist, classes, acc);
    finalize_kernel<<<1, 1, 0, stream>>>(acc, out);
}